// SelfAttention_644245094502
// MI455X (gfx1250) — compile-verified
//
#include <hip/hip_runtime.h>
#include <hip/hip_bf16.h>

// CDNA5 (gfx1250) wave32 WMMA self-attention.
// f32 path: problem is HBM-bound (~0.7GB traffic @ 23.3TB/s >> 84 GFLOP of math),
// so we keep full f32 precision and use V_WMMA_F32_16X16X4_F32.

typedef float v2f __attribute__((ext_vector_type(2)));
typedef float v8f __attribute__((ext_vector_type(8)));

__device__ __forceinline__ v8f wmma4(v2f a, v2f b, v8f c) {
  // D(16x16,f32) = A(16x4,f32) * B(4x16,f32) + C
  return __builtin_amdgcn_wmma_f32_16x16x4_f32(
      /*neg_a=*/false, a, /*neg_b=*/false, b,
      /*c_mod=*/(short)0, c, /*reuse_a=*/false, /*reuse_b=*/false);
}

// 16x64 = 16x64 @ 64x64, result written to LDS (row-major 16x64).
// A-layout (ISA 7.12.2): lane holds A(m = lane&15, k = 2*(lane>>4)+{0,1}) -> contiguous float2.
// B-layout: lane holds B(k = 2*(lane>>4)+{0,1}, n = lane&15) -> two strided dwords of W.
__device__ __forceinline__ void project16x64(const float* __restrict__ in,
                                             const float* __restrict__ W,
                                             float* __restrict__ out_lds,
                                             int lr, int half) {
  v8f acc0 = {}, acc1 = {}, acc2 = {}, acc3 = {};
  v8f acc[4] = {acc0, acc1, acc2, acc3};
#pragma unroll 4
  for (int kk = 0; kk < 16; ++kk) {
    const int kb = kk * 4 + 2 * half;
    v2f a = *(const v2f*)(in + lr * 64 + kb);
#pragma unroll
    for (int ct = 0; ct < 4; ++ct) {
      v2f b;
      b.x = W[kb * 64 + ct * 16 + lr];
      b.y = W[(kb + 1) * 64 + ct * 16 + lr];
      acc[ct] = wmma4(a, b, acc[ct]);
    }
  }
  // C/D layout: vgpr r, lane -> (m = r + 8*(lane>>4), n = lane&15)
#pragma unroll
  for (int ct = 0; ct < 4; ++ct)
#pragma unroll
    for (int r = 0; r < 8; ++r)
      out_lds[(r + 8 * half) * 64 + ct * 16 + lr] = acc[ct][r];
}

// Kernel 1: per-position fused projections + 16x16 attention + permuted store of
// the (transpose+reshape-scrambled) intermediate X into workspace.
__global__ __launch_bounds__(128) void attn_pos_kernel(
    const float* __restrict__ values, const float* __restrict__ keys,
    const float* __restrict__ query, const int* __restrict__ mask,
    const float* __restrict__ Wv, const float* __restrict__ Wk,
    const float* __restrict__ Wq, float* __restrict__ attn_out,
    float* __restrict__ X) {
  __shared__ float lds_q[4][1024];
  __shared__ float lds_k[4][1024];
  __shared__ float lds_v[4][1024];
  __shared__ float lds_e[4][256];
  __shared__ float lds_a[4][256];

  const int wave = threadIdx.x >> 5;
  const int lane = threadIdx.x & 31;
  const int half = lane >> 4;
  const int lr = lane & 15;
  const int p = blockIdx.x * 4 + wave;  // position in [0, N*L)
  const int n = p >> 12;                 // / 4096
  const int l = p & 4095;

  const float* inQ = query  + (size_t)p * 1024;
  const float* inK = keys   + (size_t)p * 1024;
  const float* inV = values + (size_t)p * 1024;

  // ---- q/k/v projections (per-head Dense(64)) ----
  project16x64(inQ, Wq, lds_q[wave], lr, half);
  project16x64(inK, Wk, lds_k[wave], lr, half);
  project16x64(inV, Wv, lds_v[wave], lr, half);
  __syncthreads();

  // ---- energy(h,g) = sum_d q[h,d] * k[g,d]  (16x16, K=64) ----
  v8f e = {};
#pragma unroll 4
  for (int kk = 0; kk < 16; ++kk) {
    const int kb = kk * 4 + 2 * half;
    v2f a = *(const v2f*)(lds_q[wave] + lr * 64 + kb);       // A(m=lr, k)
    v2f b = *(const v2f*)(lds_k[wave] + lr * 64 + kb);       // B(k, n=lr) = k[n][d]
    e = wmma4(a, b, e);
  }
#pragma unroll
  for (int r = 0; r < 8; ++r)
    lds_e[wave][(r + 8 * half) * 16 + lr] = e[r];
  __syncthreads();

  // ---- mask + softmax over g (one lane per head row) ----
  if (lane < 16) {
    const int h = lane;
    float row[16];
    float m = -3.4e38f;
#pragma unroll
    for (int g = 0; g < 16; ++g) {
      float val = lds_e[wave][h * 16 + g];
      if (mask[h * 16 + g] == 0) val = -1.0e20f;
      val *= 0.03125f;  // 1/sqrt(E), E=1024
      row[g] = val;
      m = fmaxf(m, val);
    }
    float s = 0.f;
#pragma unroll
    for (int g = 0; g < 16; ++g) { float ex = __expf(row[g] - m); row[g] = ex; s += ex; }
    const float inv = 1.f / s;
    float* aout = attn_out + (size_t)p * 256 + h * 16;
#pragma unroll
    for (int g = 0; g < 16; ++g) {
      float av = row[g] * inv;
      lds_a[wave][h * 16 + g] = av;
      aout[g] = av;
    }
  }
  __syncthreads();

  // ---- out(h,d) = attn(16x16) @ v(16x64) ----
  v8f o0 = {}, o1 = {}, o2 = {}, o3 = {};
  v8f oacc[4] = {o0, o1, o2, o3};
#pragma unroll
  for (int kk = 0; kk < 4; ++kk) {
    const int kb = kk * 4 + 2 * half;
    v2f a = *(const v2f*)(lds_a[wave] + lr * 16 + kb);
#pragma unroll
    for (int ct = 0; ct < 4; ++ct) {
      v2f b;
      b.x = lds_v[wave][kb * 64 + ct * 16 + lr];
      b.y = lds_v[wave][(kb + 1) * 64 + ct * 16 + lr];
      oacc[ct] = wmma4(a, b, oacc[ct]);
    }
  }

  // ---- scrambled store: X[n, h*256 + l/16, (l%16)*64 + d] = out[h,d] ----
  float* Xn = X + (size_t)n * 4096 * 1024;
  const int lq = l >> 4;
  const int lrm = (l & 15) * 64;
#pragma unroll
  for (int ct = 0; ct < 4; ++ct)
#pragma unroll
    for (int r = 0; r < 8; ++r) {
      const int h = r + 8 * half;
      const int d = ct * 16 + lr;
      Xn[(size_t)(h * 256 + lq) * 1024 + lrm + d] = oacc[ct][r];
    }
}

// Kernel 2: out = X(32768x1024) @ Wo(1024x1024) + bo.
// 8 waves / 256 threads per block -> 128 rows x 64 cols per block.
// Wo (4MB) lives in the 192MB L2; K-panels staged through LDS for wave sharing.
__global__ __launch_bounds__(256) void out_gemm_kernel(
    const float* __restrict__ X, const float* __restrict__ Wo,
    const float* __restrict__ bo, float* __restrict__ out) {
  __shared__ float Bp[16 * 64];

  const int wave = threadIdx.x >> 5;
  const int lane = threadIdx.x & 31;
  const int half = lane >> 4;
  const int lr = lane & 15;
  const int rowBase = blockIdx.x * 128 + wave * 16;
  const int colBase = blockIdx.y * 64;

  v8f a0 = {}, a1 = {}, a2 = {}, a3 = {};
  v8f acc[4] = {a0, a1, a2, a3};
  const float* Arow = X + (size_t)(rowBase + lr) * 1024;

  for (int kc = 0; kc < 64; ++kc) {  // K chunks of 16
    __syncthreads();
#pragma unroll
    for (int i = 0; i < 4; ++i) {
      const int t = threadIdx.x + i * 256;
      Bp[t] = Wo[(size_t)(kc * 16 + (t >> 6)) * 1024 + colBase + (t & 63)];
    }
    __syncthreads();
#pragma unroll
    for (int kk = 0; kk < 4; ++kk) {
      const int kb = kk * 4 + 2 * half;
      v2f a = *(const v2f*)(Arow + kc * 16 + kb);
#pragma unroll
      for (int ct = 0; ct < 4; ++ct) {
        v2f b;
        b.x = Bp[kb * 64 + ct * 16 + lr];
        b.y = Bp[(kb + 1) * 64 + ct * 16 + lr];
        acc[ct] = wmma4(a, b, acc[ct]);
      }
    }
  }

#pragma unroll
  for (int ct = 0; ct < 4; ++ct)
#pragma unroll
    for (int r = 0; r < 8; ++r) {
      const int row = rowBase + r + 8 * half;
      const int col = colBase + ct * 16 + lr;
      out[(size_t)row * 1024 + col] = acc[ct][r] + bo[col];
    }
}

extern "C" void kernel_launch(void* const* d_in, const int* in_sizes, int n_in,
                              void* d_out, int out_size, void* d_ws, size_t ws_size,
                              hipStream_t stream) {
  (void)in_sizes; (void)n_in; (void)out_size; (void)ws_size;
  const float* values = (const float*)d_in[0];
  const float* keys_p = (const float*)d_in[1];
  const float* query  = (const float*)d_in[2];
  const int*   mask   = (const int*)d_in[3];
  const float* Wv     = (const float*)d_in[4];
  const float* Wk     = (const float*)d_in[5];
  const float* Wq     = (const float*)d_in[6];
  const float* Wo     = (const float*)d_in[7];
  const float* bo     = (const float*)d_in[8];

  float* out  = (float*)d_out;                       // (8,4096,1024)
  float* attn = out + (size_t)8 * 4096 * 1024;       // (8,4096,16,16)
  float* X    = (float*)d_ws;                        // scrambled intermediate (8,4096,1024)

  // 32768 positions, 1 wave each, 4 waves per block
  attn_pos_kernel<<<8192, 128, 0, stream>>>(values, keys_p, query, mask,
                                            Wv, Wk, Wq, attn, X);
  // 32768x1024 rows @ 1024x1024, 128x64 per block
  out_gemm_kernel<<<dim3(256, 16), 256, 0, stream>>>(X, Wo, bo, out);
}